// holtwinter_forecasting_10479720202787
// MI455X (gfx1250) — compile-verified
//
#include <hip/hip_runtime.h>
#include <cstdint>

// Holt-Winters triple exponential smoothing scan, N=8192 series, T=2048 steps, P=24.
// Memory-bound (~128MB traffic, ~5.5us @ 23.3TB/s). One thread per series.
// CDNA5 path: TDM tensor_load_to_lds for coalesced tile staging (double buffered,
// s_wait_tensorcnt), in-register seasonal ring via period-aligned 24-col tiles,
// global_store_async_from_lds_b128 + s_wait_asynccnt for output.

#define BLK    256
#define NSER   8192
#define TLEN   2048
#define PERIOD 24
#define NTILES 86     // 85 full tiles of 24 cols (2040) + one 8-col tail
#define LASTC  8

typedef unsigned int u32x4 __attribute__((ext_vector_type(4)));
typedef int          i32x4 __attribute__((ext_vector_type(4)));
typedef int          i32x8 __attribute__((ext_vector_type(8)));

#ifndef __has_builtin
#define __has_builtin(x) 0
#endif

#if defined(__HIP_DEVICE_COMPILE__) && __has_builtin(__builtin_amdgcn_tensor_load_to_lds)
#define USE_TDM 1
#else
#define USE_TDM 0
#endif

#if __has_include(<hip/amd_detail/amd_gfx1250_TDM.h>)
#define TDM_ARITY 6   // amdgpu-toolchain / therock headers -> 6-arg builtin
#else
#define TDM_ARITY 5   // ROCm 7.2 clang-22 -> 5-arg builtin
#endif

__device__ __forceinline__ void wait_async_zero() {
#if defined(__HIP_DEVICE_COMPILE__)
#if __has_builtin(__builtin_amdgcn_s_wait_asynccnt)
  __builtin_amdgcn_s_wait_asynccnt(0);
#else
  asm volatile("s_wait_asynccnt 0x0" ::: "memory");
#endif
#endif
}

#if USE_TDM
// Issue one TDM 2D tile load: rows = 256 series (stride TLEN elems), cols elems of
// 4B each, contiguous row-major packing into LDS at lds_addr.
__device__ __forceinline__ void tdm_load_tile(const float* __restrict__ temp,
                                              unsigned lds_addr, int row_base,
                                              int t0, int cols) {
  uint64_t gaddr = (uint64_t)(uintptr_t)temp +
                   ((uint64_t)row_base * (uint64_t)TLEN + (uint64_t)t0) * 4ull;
  u32x4 g0;
  g0.x = 1u;                                                       // count=1 (valid user D#)
  g0.y = lds_addr;                                                 // LDS byte address
  g0.z = (unsigned)(gaddr & 0xffffffffull);                        // global_addr[31:0]
  g0.w = (unsigned)((gaddr >> 32) & 0x1ffffffull) | (2u << 30);    // [56:32] | type=2

  i32x8 g1;
  g1[0] = (2 << 16);                                               // data_size=4B, no pad/iter
  g1[1] = (int)((unsigned)TLEN << 16);                             // tensor_dim0 lo16
  g1[2] = (int)(((unsigned)TLEN >> 16) | ((unsigned)BLK << 16));   // dim0 hi16 | tensor_dim1 lo16
  g1[3] = (int)((unsigned)cols << 16);                             // dim1 hi16(0) | tile_dim0
  g1[4] = BLK;                                                     // tile_dim1=256, tile_dim2=0
  g1[5] = TLEN;                                                    // tensor_dim0_stride lo32
  g1[6] = 0;                                                       // stride hi | dim1_stride lo
  g1[7] = 0;

  i32x4 z4 = (i32x4)0;
#if TDM_ARITY == 6
  i32x8 z8 = (i32x8)0;
  __builtin_amdgcn_tensor_load_to_lds(g0, g1, z4, z4, z8, 0);
#else
  __builtin_amdgcn_tensor_load_to_lds(g0, g1, z4, z4, 0);
#endif
}
#else
// Fallback: cooperative coalesced load (96B runs per row).
__device__ __forceinline__ void coop_load_tile(float* dst, const float* __restrict__ temp,
                                               int row_base, int t0, int cols, int tid) {
  const int total = BLK * cols;
  for (int f = tid; f < total; f += BLK) {
    int r = f / cols;
    int c = f - r * cols;
    dst[r * cols + c] = temp[(size_t)(row_base + r) * TLEN + (size_t)(t0 + c)];
  }
}
#endif

// Per-thread Holt-Winters update over one tile. Phase of column i is exactly i
// (tiles are period-aligned), so the seasonal buffer s[24] stays in registers.
// Output overwrites the input slot in LDS (each slot read once by its owner).
template <int COLS>
__device__ __forceinline__ void compute_tile(float* tile, int tid, int t0,
                                             float& L, float& b, float (&s)[PERIOD],
                                             float a, float bt, float g) {
  float* row = tile + tid * COLS;
#pragma unroll
  for (int i = 0; i < COLS; ++i) {
    const int t = t0 + i;
    const float x = row[i];
    float outv;
    if (t == 0) {
      outv = (L + b) * s[0];                 // out[:,0] = (L0+b0)*S0[:,0]
    } else {
      const float sr = s[i];                 // phase == i
      const float Ln = a * (x - sr) + (1.0f - a) * (L + b);
      const float bn = bt * (Ln - L) + (1.0f - bt) * b;
      const float sn = g * (x - Ln) + (1.0f - g) * sr;
      s[i] = sn;
      L = Ln;
      b = bn;
      outv = (Ln + bn) * sn;
    }
    row[i] = outv;                           // in-place result
  }
}

// Async LDS -> global store, 16B chunks (COLS is a multiple of 4).
template <int COLS>
__device__ __forceinline__ void store_tile(unsigned lds_base, float* __restrict__ out,
                                           int row_base, int t0, int tid) {
#if defined(__HIP_DEVICE_COMPILE__)
  constexpr int CPR = COLS / 4;              // 16B chunks per row
#pragma unroll
  for (int it = 0; it < CPR; ++it) {
    const int f = it * BLK + tid;
    const int r = f / CPR;
    const int c = f - r * CPR;
    const unsigned lsrc = lds_base + (unsigned)(r * (COLS * 4) + c * 16);
    const unsigned voff =
        (unsigned)(((unsigned)(row_base + r) * (unsigned)TLEN + (unsigned)t0) * 4u +
                   (unsigned)c * 16u);
    asm volatile("global_store_async_from_lds_b128 %0, %1, %2"
                 :
                 : "v"(voff), "v"(lsrc), "s"(out)
                 : "memory");
  }
#else
  (void)lds_base; (void)out; (void)row_base; (void)t0; (void)tid;
#endif
}

__global__ __launch_bounds__(BLK) void holtwinter_scan_kernel(
    const float* __restrict__ temp, const float* __restrict__ alpha,
    const float* __restrict__ beta, const float* __restrict__ gamma,
    const float* __restrict__ L0, const float* __restrict__ b0,
    const float* __restrict__ S0, float* __restrict__ out) {
  __shared__ float tiles[2][BLK * PERIOD];   // 2 x 24KB double buffer

  const int tid = (int)threadIdx.x;
  const int row_base = (int)blockIdx.x * BLK;
  const int n = row_base + tid;

  const float a = alpha[0];
  const float bt = beta[0];
  const float g = gamma[0];

  float L = L0[n];
  float b = b0[n];
  float s[PERIOD];
#pragma unroll
  for (int j = 0; j < PERIOD; ++j) s[j] = S0[n * PERIOD + j];

  const unsigned lds0 = (unsigned)(uintptr_t)&tiles[0][0];
  const unsigned lds1 = (unsigned)(uintptr_t)&tiles[1][0];

  // Prologue: stage tile 0.
#if USE_TDM
  if (tid < 32) tdm_load_tile(temp, lds0, row_base, 0, PERIOD);
#else
  coop_load_tile(&tiles[0][0], temp, row_base, 0, PERIOD, tid);
#endif

  for (int k = 0; k < NTILES; ++k) {
    const int cur = k & 1;
    const int nxt = cur ^ 1;
    const int t0 = k * PERIOD;
    const unsigned lds_cur = cur ? lds1 : lds0;
    const unsigned lds_nxt = cur ? lds0 : lds1;

    // Buffer 'nxt' is about to be refilled: drain our async stores that read it.
    wait_async_zero();
    __syncthreads();

#if USE_TDM
    if (tid < 32) {                          // wave 0 only (uniform branch)
      if (k + 1 < NTILES) {
        tdm_load_tile(temp, lds_nxt, row_base, (k + 1) * PERIOD,
                      (k + 1 == NTILES - 1) ? LASTC : PERIOD);
        __builtin_amdgcn_s_wait_tensorcnt(1);  // tile k complete (in-order)
      } else {
        __builtin_amdgcn_s_wait_tensorcnt(0);
      }
    }
#else
    if (k + 1 < NTILES)
      coop_load_tile(&tiles[nxt][0], temp, row_base, (k + 1) * PERIOD,
                     (k + 1 == NTILES - 1) ? LASTC : PERIOD, tid);
#endif
    __syncthreads();
    asm volatile("" ::: "memory");           // tile data materialized behind compiler's back

    if (k < NTILES - 1)
      compute_tile<PERIOD>(&tiles[cur][0], tid, t0, L, b, s, a, bt, g);
    else
      compute_tile<LASTC>(&tiles[cur][0], tid, t0, L, b, s, a, bt, g);

    __syncthreads();

    if (k < NTILES - 1)
      store_tile<PERIOD>(lds_cur, out, row_base, t0, tid);
    else
      store_tile<LASTC>(lds_cur, out, row_base, t0, tid);
  }
  // s_endpgm performs an implicit wait-idle for outstanding async stores.
}

extern "C" void kernel_launch(void* const* d_in, const int* in_sizes, int n_in,
                              void* d_out, int out_size, void* d_ws, size_t ws_size,
                              hipStream_t stream) {
  (void)in_sizes; (void)n_in; (void)out_size; (void)d_ws; (void)ws_size;
  const float* temp  = (const float*)d_in[0];
  const float* alpha = (const float*)d_in[1];
  const float* beta  = (const float*)d_in[2];
  const float* gamma = (const float*)d_in[3];
  const float* L0    = (const float*)d_in[4];
  const float* b0    = (const float*)d_in[5];
  const float* S0    = (const float*)d_in[6];
  float* out = (float*)d_out;

  dim3 grid(NSER / BLK);
  dim3 block(BLK);
  hipLaunchKernelGGL(holtwinter_scan_kernel, grid, block, 0, stream,
                     temp, alpha, beta, gamma, L0, b0, S0, out);
}